// PatchEmbedding_47828755808654
// MI455X (gfx1250) — compile-verified
//
#include <hip/hip_runtime.h>

typedef __attribute__((ext_vector_type(2))) float v2f;
typedef __attribute__((ext_vector_type(8))) float v8f;

#define IMG_H 96
#define IMG_W 96
#define CCH   64
#define NB    32
#define HP    89
#define WPOS  89
#define LPATCH (HP * WPOS)   // 7921
#define DM    256
#define KDIM  64
#define LDS_ELEMS 800        // 8*96 staged + zero tail (max read idx 774)

// ---------------- Kernel 1: channel mean ----------------
__global__ void __launch_bounds__(256)
pe_mean_kernel(const float* __restrict__ x, float* __restrict__ xm) {
    int idx = blockIdx.x * 256 + threadIdx.x;        // (b, pixel)
    if (idx >= NB * IMG_H * IMG_W) return;
    int b = idx / (IMG_H * IMG_W);
    int p = idx - b * (IMG_H * IMG_W);
    const float* base = x + (size_t)b * CCH * IMG_H * IMG_W + p;
    float acc = 0.f;
#pragma unroll 16
    for (int c = 0; c < CCH; ++c)
        acc += base[(size_t)c * (IMG_H * IMG_W)];
    xm[idx] = acc * (1.0f / 64.0f);
}

// ---------------- Kernel 2: patch-GEMM via f32 WMMA ----------------
// grid = (89 patch-rows, 32 batches), block = 512 (16 waves)
// wave w -> N-tile d in [16w, 16w+16); loops 6 M-tiles of patch columns.
__global__ void __launch_bounds__(512)
pe_wmma_kernel(const float* __restrict__ xm,
               const float* __restrict__ Wt,     // (256, 64) row-major
               const float* __restrict__ bias,   // (256,)
               float* __restrict__ out) {        // (32, 7921, 256)
    __shared__ float lds[LDS_ELEMS];

    const int i   = blockIdx.x;   // patch row 0..88
    const int b   = blockIdx.y;   // batch
    const int tid = threadIdx.x;

    // Stage image rows i..i+7 (contiguous 768 floats); zero-fill tail.
    const float* src = xm + ((size_t)b * IMG_H + i) * IMG_W;
    for (int t = tid; t < LDS_ELEMS; t += 512)
        lds[t] = (t < 8 * IMG_W) ? src[t] : 0.0f;
    __syncthreads();

    const int lane = tid & 31;
    const int wv   = tid >> 5;        // 0..15 -> N-tile
    const int n0   = wv * 16;
    const int m15  = lane & 15;
    const int half = lane >> 4;       // 0 or 1

    // B fragment: Wt^T slice. B(k, n) = Wt[n, k]; per ISA f32 layout
    // VGPR0 holds K = 0 + 2*half, VGPR1 holds K = 1 + 2*half per 4-step.
    v2f bfrag[16];
    const v2f* wrow = (const v2f*)(Wt + (size_t)(n0 + m15) * KDIM + 2 * half);
#pragma unroll
    for (int s = 0; s < 16; ++s)
        bfrag[s] = wrow[2 * s];       // (Wt[n,4s+2h], Wt[n,4s+2h+1])

    const float bv = bias[n0 + m15];  // D-layout column n = lane&15

    for (int mt = 0; mt < 6; ++mt) {
        const int j0 = mt * 16;
        const int jA = j0 + m15;      // A-row = patch column j (may exceed 88)

        v8f acc0 = {};
        v8f acc1 = {};
#pragma unroll
        for (int s = 0; s < 16; ++s) {
            // K-values 4s..4s+3 this step; this lane supplies k0, k0+1.
            const int k0 = 4 * s + 2 * half;
            const int r  = k0 >> 3;           // window row  (c stays in-row:
            const int c  = k0 & 7;            //  c in {0,2,4,6})
            v2f a;
            a.x = lds[r * IMG_W + jA + c];
            a.y = lds[r * IMG_W + jA + c + 1];
            if (s & 1)
                acc1 = __builtin_amdgcn_wmma_f32_16x16x4_f32(
                    false, a, false, bfrag[s], (short)0, acc1, false, false);
            else
                acc0 = __builtin_amdgcn_wmma_f32_16x16x4_f32(
                    false, a, false, bfrag[s], (short)0, acc0, false, false);
        }

        // Store 16x16 tile: D vgpr v -> patch col j0 + v + 8*half, col n0+m15.
        const int jbase = j0 + 8 * half;
#pragma unroll
        for (int v = 0; v < 8; ++v) {
            const int j = jbase + v;
            if (j < WPOS) {
                size_t off = (((size_t)b * LPATCH) + (size_t)i * WPOS + j) * DM
                             + (size_t)(n0 + m15);
                out[off] = acc0[v] + acc1[v] + bv;
            }
        }
    }
}

extern "C" void kernel_launch(void* const* d_in, const int* in_sizes, int n_in,
                              void* d_out, int out_size, void* d_ws, size_t ws_size,
                              hipStream_t stream) {
    const float* x    = (const float*)d_in[0];
    const float* Wt   = (const float*)d_in[1];
    const float* bias = (const float*)d_in[2];
    float* out = (float*)d_out;
    float* xm  = (float*)d_ws;   // 32*96*96 floats = 1.18 MB scratch

    const int npix = NB * IMG_H * IMG_W;
    pe_mean_kernel<<<dim3((npix + 255) / 256), dim3(256), 0, stream>>>(x, xm);

    pe_wmma_kernel<<<dim3(HP, NB), dim3(512), 0, stream>>>(xm, Wt, bias, out);
}